// LuongAttnDecoderRNN_29996051595407
// MI455X (gfx1250) — compile-verified
//
#include <hip/hip_runtime.h>
#include <math.h>

typedef __attribute__((ext_vector_type(2))) float v2f;
typedef __attribute__((ext_vector_type(8))) float v8f;

#define BATCH   64
#define LSEQ    2048
#define HDIM    1024
#define VOCABN  32000
#define CHUNKS  8
#define ROWS_PER_BLOCK (LSEQ / CHUNKS)      // 256
#define ROWS_PER_WAVE  (ROWS_PER_BLOCK / 8) // 32

// ---------------------------------------------------------------------------
// Embedding gather: x[b][:] = table[idx[b]][:]
// ---------------------------------------------------------------------------
__global__ void embed_kernel(const int* __restrict__ idx,
                             const float* __restrict__ table,
                             float* __restrict__ x) {
  const int b = blockIdx.x;
  const int token = idx[b];
  const float* src = table + (size_t)token * HDIM;
  float* dst = x + (size_t)b * HDIM;
  for (int e = threadIdx.x; e < HDIM; e += blockDim.x) dst[e] = src[e];
}

// ---------------------------------------------------------------------------
// NT GEMM via V_WMMA_F32_16X16X4_F32:
//   C[M][N] = act( A[M][K] @ B[N][K]^T + bias[N] )
// One wave per 16x16 output tile; both A and B are K-contiguous row-major.
// fp32 A-fragment layout (ISA 7.12.2): lanes 0-15 -> (M=r, K=k..k+1),
// lanes 16-31 -> (M=r, K=k+2..k+3); B symmetric with N in place of M.
// C/D: lane r holds column N=r; VGPR v holds row M=v (lanes 0-15) or
// M=v+8 (lanes 16-31).
// ---------------------------------------------------------------------------
__global__ __launch_bounds__(256) void gemm_nt_wmma(
    const float* __restrict__ A, const float* __restrict__ B,
    const float* __restrict__ bias, float* __restrict__ C,
    int M, int N, int K, int act) {
  const int wave = threadIdx.x >> 5;
  const int lane = threadIdx.x & 31;
  const int hf   = lane >> 4;   // which half-wave (K pair select)
  const int r    = lane & 15;
  const int mtiles = M >> 4;
  const int ntiles = N >> 4;
  const int tile = blockIdx.x * 8 + wave;
  if (tile >= mtiles * ntiles) return;   // wave-uniform; EXEC stays full
  const int mt = tile % mtiles;
  const int nt = tile / mtiles;

  const float* ap = A + (size_t)(mt * 16 + r) * K + 2 * hf;
  const float* bp = B + (size_t)(nt * 16 + r) * K + 2 * hf;

  v8f acc = {};
#pragma unroll 4
  for (int k = 0; k < K; k += 4) {
    v2f av = *(const v2f*)(ap + k);
    v2f bv = *(const v2f*)(bp + k);
    acc = __builtin_amdgcn_wmma_f32_16x16x4_f32(
        /*neg_a=*/false, av, /*neg_b=*/false, bv,
        /*c_mod=*/(short)0, acc, /*reuse_a=*/false, /*reuse_b=*/false);
  }

  const int col = nt * 16 + r;
  const float bb = bias[col];
#pragma unroll
  for (int v = 0; v < 8; ++v) {
    float val = acc[v] + bb;
    if (act == 1) val = tanhf(val);
    C[(size_t)(mt * 16 + v + 8 * hf) * N + col] = val;
  }
}

// ---------------------------------------------------------------------------
// GRU gate pointwise: h = (1-z)*n + z*h_prev
// gi/gh are [B][3H] in (r,z,n) chunk order.
// ---------------------------------------------------------------------------
__global__ void gru_pointwise(const float* __restrict__ gi,
                              const float* __restrict__ gh,
                              const float* __restrict__ hprev,
                              float* __restrict__ hout,
                              float* __restrict__ hout2) {
  const int i = blockIdx.x * blockDim.x + threadIdx.x;
  if (i >= BATCH * HDIM) return;
  const int b = i >> 10;
  const int j = i & (HDIM - 1);
  const size_t base = (size_t)b * 3 * HDIM + j;
  const float ir = gi[base], iz = gi[base + HDIM], inn = gi[base + 2 * HDIM];
  const float hr = gh[base], hz = gh[base + HDIM], hn  = gh[base + 2 * HDIM];
  const float rg = 1.f / (1.f + __expf(-(ir + hr)));
  const float zg = 1.f / (1.f + __expf(-(iz + hz)));
  const float ng = tanhf(inn + rg * hn);
  const float h  = (1.f - zg) * ng + zg * hprev[i];
  hout[i]  = h;
  hout2[i] = h;
}

// ---------------------------------------------------------------------------
// Attention pass 1 (flash-style, single read of encoder_outputs):
// block = (batch b, chunk of 256 L rows); 8 waves x 32 rows each.
// Each wave: load enc row into regs, score = shuffle-reduced dot with h1,
// online-softmax rescale of per-lane context accumulator (32 elems/lane).
// Wave partials (m, s, ctx) combined through LDS into one block partial.
// Raw scores stored for the attn_weights output.
// ---------------------------------------------------------------------------
__global__ __launch_bounds__(256) void attn_pass1(
    const float* __restrict__ h1, const float* __restrict__ enc,
    float* __restrict__ scores, float* __restrict__ pm,
    float* __restrict__ ps, float* __restrict__ pctx) {
  const int b     = blockIdx.x / CHUNKS;
  const int chunk = blockIdx.x % CHUNKS;
  const int lane  = threadIdx.x & 31;
  const int wave  = threadIdx.x >> 5;

  float4 hreg[8];
  const float* hb = h1 + (size_t)b * HDIM;
#pragma unroll
  for (int i = 0; i < 8; ++i)
    hreg[i] = *(const float4*)(hb + i * 128 + lane * 4);

  float m = -3.0e38f, s = 0.f;
  float4 ctx[8];
#pragma unroll
  for (int i = 0; i < 8; ++i) ctx[i] = make_float4(0.f, 0.f, 0.f, 0.f);

  const int row0 = chunk * ROWS_PER_BLOCK + wave * ROWS_PER_WAVE;
  for (int rr = 0; rr < ROWS_PER_WAVE; ++rr) {
    const int l = row0 + rr;
    const float* erow = enc + ((size_t)b * LSEQ + l) * HDIM;
    __builtin_prefetch(erow + HDIM, 0, 0);  // global_prefetch_b8, next row
    float4 ereg[8];
    float dot = 0.f;
#pragma unroll
    for (int i = 0; i < 8; ++i) {
      ereg[i] = *(const float4*)(erow + i * 128 + lane * 4);
      dot += hreg[i].x * ereg[i].x + hreg[i].y * ereg[i].y +
             hreg[i].z * ereg[i].z + hreg[i].w * ereg[i].w;
    }
#pragma unroll
    for (int off = 16; off > 0; off >>= 1) dot += __shfl_xor(dot, off, 32);
    if (lane == 0) scores[(size_t)b * LSEQ + l] = dot;

    const float mn    = fmaxf(m, dot);
    const float scale = __expf(m - mn);
    const float w     = __expf(dot - mn);
    s = s * scale + w;
#pragma unroll
    for (int i = 0; i < 8; ++i) {
      ctx[i].x = ctx[i].x * scale + w * ereg[i].x;
      ctx[i].y = ctx[i].y * scale + w * ereg[i].y;
      ctx[i].z = ctx[i].z * scale + w * ereg[i].z;
      ctx[i].w = ctx[i].w * scale + w * ereg[i].w;
    }
    m = mn;
  }

  __shared__ __align__(16) float sctx[8 * HDIM];  // 32 KB (of 320 KB/WGP)
  __shared__ float sm[8], ss[8];
#pragma unroll
  for (int i = 0; i < 8; ++i)
    *(float4*)(sctx + wave * HDIM + i * 128 + lane * 4) = ctx[i];
  if (lane == 0) { sm[wave] = m; ss[wave] = s; }
  __syncthreads();

  float mb = -3.0e38f;
  for (int w = 0; w < 8; ++w) mb = fmaxf(mb, sm[w]);
  if (threadIdx.x == 0) {
    float sb = 0.f;
    for (int w = 0; w < 8; ++w) sb += ss[w] * __expf(sm[w] - mb);
    pm[blockIdx.x] = mb;
    ps[blockIdx.x] = sb;
  }
  for (int e = threadIdx.x; e < HDIM; e += blockDim.x) {
    float acc = 0.f;
    for (int w = 0; w < 8; ++w) acc += sctx[w * HDIM + e] * __expf(sm[w] - mb);
    pctx[(size_t)blockIdx.x * HDIM + e] = acc;
  }
}

// ---------------------------------------------------------------------------
// Attention combine: merge CHUNKS block partials per batch; emit context and
// the concat buffer [h1 | context], plus final (m*, s*) for weight normalize.
// ---------------------------------------------------------------------------
__global__ void attn_combine(const float* __restrict__ pm,
                             const float* __restrict__ ps,
                             const float* __restrict__ pctx,
                             const float* __restrict__ h1,
                             float* __restrict__ fm, float* __restrict__ fs,
                             float* __restrict__ concat) {
  const int b = blockIdx.x;
  float ms = -3.0e38f;
  for (int c = 0; c < CHUNKS; ++c) ms = fmaxf(ms, pm[b * CHUNKS + c]);
  float ssum = 0.f;
  for (int c = 0; c < CHUNKS; ++c)
    ssum += ps[b * CHUNKS + c] * __expf(pm[b * CHUNKS + c] - ms);
  if (threadIdx.x == 0) { fm[b] = ms; fs[b] = ssum; }
  const float inv = 1.f / ssum;
  for (int e = threadIdx.x; e < HDIM; e += blockDim.x) {
    float acc = 0.f;
    for (int c = 0; c < CHUNKS; ++c)
      acc += pctx[((size_t)b * CHUNKS + c) * HDIM + e] *
             __expf(pm[b * CHUNKS + c] - ms);
    concat[(size_t)b * 2 * HDIM + HDIM + e] = acc * inv;
    concat[(size_t)b * 2 * HDIM + e]        = h1[(size_t)b * HDIM + e];
  }
}

// ---------------------------------------------------------------------------
// attn_weights[b][l] = exp(score - m*) / s*
// ---------------------------------------------------------------------------
__global__ void attn_weights_kernel(const float* __restrict__ scores,
                                    const float* __restrict__ fm,
                                    const float* __restrict__ fs,
                                    float* __restrict__ out_attn) {
  const int i = blockIdx.x * blockDim.x + threadIdx.x;
  if (i >= BATCH * LSEQ) return;
  const int b = i / LSEQ;
  out_attn[i] = __expf(scores[i] - fm[b]) / fs[b];
}

// ---------------------------------------------------------------------------
// Host-side orchestration (all on `stream`, graph-capture safe)
// ---------------------------------------------------------------------------
extern "C" void kernel_launch(void* const* d_in, const int* in_sizes, int n_in,
                              void* d_out, int out_size, void* d_ws,
                              size_t ws_size, hipStream_t stream) {
  const int*   input_seq = (const int*)  d_in[0];
  const float* last_hid  = (const float*)d_in[1];
  const float* enc       = (const float*)d_in[2];
  const float* embed     = (const float*)d_in[3];
  const float* W_ih0     = (const float*)d_in[4];
  const float* W_hh0     = (const float*)d_in[5];
  const float* b_ih0     = (const float*)d_in[6];
  const float* b_hh0     = (const float*)d_in[7];
  const float* W_ih1     = (const float*)d_in[8];
  const float* W_hh1     = (const float*)d_in[9];
  const float* b_ih1     = (const float*)d_in[10];
  const float* b_hh1     = (const float*)d_in[11];
  const float* Wc        = (const float*)d_in[12];
  const float* bc        = (const float*)d_in[13];
  const float* Wout      = (const float*)d_in[14];
  const float* bout      = (const float*)d_in[15];

  float* out        = (float*)d_out;
  float* out_hidden = out + (size_t)BATCH * VOCABN;          // 2,048,000
  float* out_attn   = out_hidden + (size_t)2 * BATCH * HDIM; // +131,072

  // workspace layout (floats), ~5.8 MB total
  float* ws      = (float*)d_ws;
  float* x       = ws;                  // 65536
  float* gi      = ws + 65536;          // 196608
  float* gh      = ws + 262144;         // 196608
  float* h0      = ws + 458752;         // 65536
  float* h1      = ws + 524288;         // 65536
  float* scores  = ws + 589824;         // 131072
  float* pm      = ws + 720896;         // 512
  float* ps      = ws + 721408;         // 512
  float* pctx    = ws + 721920;         // 524288
  float* fm      = ws + 1246208;        // 64
  float* fs      = ws + 1246272;        // 64
  float* concat  = ws + 1246336;        // 131072
  float* co      = ws + 1377408;        // 65536

  const float* lh0 = last_hid;                       // layer 0 hidden
  const float* lh1 = last_hid + (size_t)BATCH * HDIM; // layer 1 hidden

  auto gemm_blocks = [](int M, int N) { return ((M >> 4) * (N >> 4) + 7) / 8; };

  // 1) embedding gather
  embed_kernel<<<BATCH, 256, 0, stream>>>(input_seq, embed, x);

  // 2) GRU layer 0
  gemm_nt_wmma<<<gemm_blocks(BATCH, 3 * HDIM), 256, 0, stream>>>(
      x, W_ih0, b_ih0, gi, BATCH, 3 * HDIM, HDIM, 0);
  gemm_nt_wmma<<<gemm_blocks(BATCH, 3 * HDIM), 256, 0, stream>>>(
      lh0, W_hh0, b_hh0, gh, BATCH, 3 * HDIM, HDIM, 0);
  gru_pointwise<<<(BATCH * HDIM + 255) / 256, 256, 0, stream>>>(
      gi, gh, lh0, h0, out_hidden);

  // 3) GRU layer 1
  gemm_nt_wmma<<<gemm_blocks(BATCH, 3 * HDIM), 256, 0, stream>>>(
      h0, W_ih1, b_ih1, gi, BATCH, 3 * HDIM, HDIM, 0);
  gemm_nt_wmma<<<gemm_blocks(BATCH, 3 * HDIM), 256, 0, stream>>>(
      lh1, W_hh1, b_hh1, gh, BATCH, 3 * HDIM, HDIM, 0);
  gru_pointwise<<<(BATCH * HDIM + 255) / 256, 256, 0, stream>>>(
      gi, gh, lh1, h1, out_hidden + (size_t)BATCH * HDIM);

  // 4) attention (single streaming pass over encoder_outputs)
  attn_pass1<<<BATCH * CHUNKS, 256, 0, stream>>>(h1, enc, scores, pm, ps, pctx);
  attn_combine<<<BATCH, 256, 0, stream>>>(pm, ps, pctx, h1, fm, fs, concat);
  attn_weights_kernel<<<(BATCH * LSEQ + 255) / 256, 256, 0, stream>>>(
      scores, fm, fs, out_attn);

  // 5) concat projection with tanh
  gemm_nt_wmma<<<gemm_blocks(BATCH, HDIM), 256, 0, stream>>>(
      concat, Wc, bc, co, BATCH, HDIM, 2 * HDIM, 1);

  // 6) vocab projection
  gemm_nt_wmma<<<gemm_blocks(BATCH, VOCABN), 256, 0, stream>>>(
      co, Wout, bout, out, BATCH, VOCABN, HDIM, 0);
}